// MeshfreeKAN1D_12996571038257
// MI455X (gfx1250) — compile-verified
//
#include <hip/hip_runtime.h>
#include <math.h>

typedef __attribute__((ext_vector_type(2))) float v2f;
typedef __attribute__((ext_vector_type(8))) float v8f;

constexpr int   NUM_BASIS  = 7;
constexpr int   HIDDEN     = 16;
constexpr float GMIN       = -1.5f;
constexpr float H          = 0.5f;     // grid spacing
constexpr float INVH       = 2.0f;     // 1/h
constexpr float INV_RADIUS = 50.0f;    // 1/0.02
constexpr int   NN_MAX     = 512;
constexpr int   THREADS    = 256;
constexpr int   WAVES      = THREADS / 32;

__device__ __forceinline__ float hatb(float v, float gk) {
    // relu(1 - |v - gk| / h)
    return fmaxf(0.0f, 1.0f - fabsf(v - gk) * INVH);
}

__global__ __launch_bounds__(THREADS)
void kan_row_kernel(const float* __restrict__ x,
                    const float* __restrict__ nodes,
                    const float* __restrict__ W1,
                    const float* __restrict__ W2,
                    float* __restrict__ out,
                    int Ne, int Nn)
{
    __shared__ float s_nodes[NN_MAX];
    __shared__ float s_phiw[NN_MAX];
    __shared__ float s_dist[NN_MAX];
    __shared__ float s_W2[HIDDEN * NUM_BASIS];
    __shared__ float s_red[THREADS];
    __shared__ int   s_redi[THREADS];
    __shared__ float s_selD[3];
    __shared__ int   s_selI[3];

    const int tid  = threadIdx.x;
    const int row  = blockIdx.x;
    const int lane = tid & 31;
    const int wave = tid >> 5;
    const int n16  = lane & 15;   // M for A-operand, N for B/D
    const int hi   = lane >> 4;   // K-half selector

    for (int j = tid; j < Nn; j += THREADS)               s_nodes[j] = nodes[j];
    for (int j = tid; j < HIDDEN * NUM_BASIS; j += THREADS) s_W2[j]  = W2[j];

    const float xi = x[row];

    // Loop-invariant A operands: W1 striped per 16x4 f32 A layout.
    // WMMA #1 covers K = 0..3, WMMA #2 covers K = 4..7 (K=7 zero-padded).
    const int m = n16;
    v2f a1, a2;
    a1.x = W1[m * NUM_BASIS + 2 * hi + 0];
    a1.y = W1[m * NUM_BASIS + 2 * hi + 1];
    a2.x = W1[m * NUM_BASIS + 4 + 2 * hi];
    a2.y = hi ? 0.0f : W1[m * NUM_BASIS + 5];

    __syncthreads();

    const int nblk = Nn >> 4;                 // 16 pairs per wave-step
    for (int blk = wave; blk < nblk; blk += WAVES) {
        const int   j    = (blk << 4) + n16;
        const float nd   = s_nodes[j];
        const float diff = xi - nd;
        const float xr   = diff * INV_RADIUS;

        // B operand: b1 hat-basis values, B layout mirrors A (N across lanes).
        v2f b1v, b2v;
        {
            const float g0 = GMIN + H * (float)(2 * hi);
            b1v.x = hatb(xr, g0);
            b1v.y = hatb(xr, g0 + H);
            const float g4 = GMIN + H * (float)(4 + 2 * hi);
            b2v.x = hatb(xr, g4);
            b2v.y = hi ? 0.0f : hatb(xr, g4 + H);   // K=7 pad
        }

        // hid[16][16 pairs] = W1(16x7) x b1T(7x16), fp32 WMMA, chained accumulate
        v8f c = {};
        c = __builtin_amdgcn_wmma_f32_16x16x4_f32(false, a1, false, b1v,
                                                  (short)0, c, false, false);
        c = __builtin_amdgcn_wmma_f32_16x16x4_f32(false, a2, false, b2v,
                                                  (short)0, c, false, false);

        // Stage 2: each lane owns hidden rows m = 8*hi + r of pair n = lane&15.
        float acc = 0.0f;
#pragma unroll
        for (int r = 0; r < 8; ++r) {
            const float hv = c[r];
            const float* w2p = &s_W2[(8 * hi + r) * NUM_BASIS];
#pragma unroll
            for (int k = 0; k < NUM_BASIS; ++k) {
                acc = fmaf(w2p[k], hatb(hv, GMIN + H * (float)k), acc);
            }
        }
        // Combine the two M-halves of this pair (lanes l and l^16).
        acc += __shfl_xor(acc, 16, 32);

        // softplus (numerically stable)
        const float sp = fmaxf(acc, 0.0f) + log1pf(expf(-fabsf(acc)));

        // cubic window
        const float dist = fabsf(diff);
        const float q    = dist * INV_RADIUS;
        float wnd;
        if (q <= 0.5f)
            wnd = 2.0f / 3.0f + q * q * (4.0f * q - 4.0f);
        else if (q <= 1.0f)
            wnd = 4.0f / 3.0f - 4.0f * q + 4.0f * q * q - (4.0f / 3.0f) * q * q * q;
        else
            wnd = 0.0f;

        if (hi == 0) {
            s_phiw[j] = sp * wnd;
            s_dist[j] = dist;
        }
    }
    __syncthreads();

    // Row sum of phi_w
    {
        float s = 0.0f;
        for (int j2 = tid; j2 < Nn; j2 += THREADS) s += s_phiw[j2];
        s_red[tid] = s;
        __syncthreads();
        for (int off = THREADS / 2; off > 0; off >>= 1) {
            if (tid < off) s_red[tid] += s_red[tid + off];
            __syncthreads();
        }
    }
    const float phi_sum = s_red[0];
    __syncthreads();

    // Top-3 nearest nodes (three successive min-reductions, tie-break low idx)
    for (int round = 0; round < 3; ++round) {
        float bd = 1e30f;
        int   bi = 0x7fffffff;
        for (int j2 = tid; j2 < Nn; j2 += THREADS) {
            bool skip = false;
            for (int r = 0; r < round; ++r)
                if (s_selI[r] == j2) skip = true;
            const float d = s_dist[j2];
            if (!skip && (d < bd || (d == bd && j2 < bi))) { bd = d; bi = j2; }
        }
        s_red[tid]  = bd;
        s_redi[tid] = bi;
        __syncthreads();
        for (int off = THREADS / 2; off > 0; off >>= 1) {
            if (tid < off) {
                const float d2 = s_red[tid + off];
                const int   i2 = s_redi[tid + off];
                if (d2 < s_red[tid] || (d2 == s_red[tid] && i2 < s_redi[tid])) {
                    s_red[tid] = d2; s_redi[tid] = i2;
                }
            }
            __syncthreads();
        }
        if (tid == 0) { s_selD[round] = s_red[0]; s_selI[round] = s_redi[0]; }
        __syncthreads();
    }

    // KNN softmax weights (selD is ascending by construction)
    const bool orphan = fabsf(phi_sum) < 1e-14f;
    float w0, w1, w2;
    {
        const float alpha = 20.0f * INV_RADIUS;   // 20 / RADIUS = 1000
        const float dmin  = s_selD[0];
        w0 = expf(-alpha * (s_selD[0] - dmin));
        w1 = expf(-alpha * (s_selD[1] - dmin));
        w2 = expf(-alpha * (s_selD[2] - dmin));
        const float ws = w0 + w1 + w2;
        w0 /= ws; w1 /= ws; w2 /= ws;
    }

    const float inv = 1.0f / (phi_sum + 1e-12f);
    for (int j2 = tid; j2 < Nn; j2 += THREADS) {
        float val;
        if (orphan) {
            val = 0.0f;
            if      (j2 == s_selI[0]) val = w0;
            else if (j2 == s_selI[1]) val = w1;
            else if (j2 == s_selI[2]) val = w2;
        } else {
            val = s_phiw[j2] * inv;
        }
        out[(size_t)row * (size_t)Nn + j2] = val;
    }
}

extern "C" void kernel_launch(void* const* d_in, const int* in_sizes, int n_in,
                              void* d_out, int out_size, void* d_ws, size_t ws_size,
                              hipStream_t stream) {
    const float* x     = (const float*)d_in[0];
    const float* nodes = (const float*)d_in[1];
    const float* W1    = (const float*)d_in[2];
    const float* W2    = (const float*)d_in[3];
    float* out = (float*)d_out;
    const int Ne = in_sizes[0];
    const int Nn = in_sizes[1];
    kan_row_kernel<<<Ne, THREADS, 0, stream>>>(x, nodes, W1, W2, out, Ne, Nn);
}